// HardTripletLoss_50818053046377
// MI455X (gfx1250) — compile-verified
//
#include <hip/hip_runtime.h>
#include <hip/hip_bf16.h>
#include <stdint.h>

typedef __attribute__((ext_vector_type(2))) float v2f;
typedef __attribute__((ext_vector_type(8))) float v8f;
typedef __attribute__((ext_vector_type(4))) int   v4i;
typedef __attribute__((address_space(1))) v4i as1_v4i;
typedef __attribute__((address_space(3))) v4i as3_v4i;

#define N_ROWS 8192
#define DIM    128
#define MARGIN 0.5f
#define BIG    1e30f
#define TILE   16
#define NTILES (N_ROWS / TILE)     // 512
#define WAVES  8
#define THREADS (WAVES * 32)       // 256
#define JGROUPS 8
#define TILES_PER_GROUP (NTILES / JGROUPS)   // 64 j-tiles per block
#define ITER (TILES_PER_GROUP / 2)           // 32 iterations, 2 tiles each
#define BROW_STRIDE 132            // 128 + 4 floats pad -> bank = (4*row+col)%64
#define BUF_FLOATS (32 * BROW_STRIDE)

#ifndef __has_builtin
#define __has_builtin(x) 0
#endif
#if __has_builtin(__builtin_amdgcn_global_load_async_to_lds_b128) && \
    __has_builtin(__builtin_amdgcn_s_wait_asynccnt)
#define USE_ASYNC_COPY 1
#else
#define USE_ASYNC_COPY 0
#endif

// ---------------------------------------------------------------------------
// Kernel A: per-row squared norms. One row per wave; lane loads float4.
// ---------------------------------------------------------------------------
__global__ __launch_bounds__(256) void sqnorm_kernel(const float* __restrict__ X,
                                                     float* __restrict__ sq) {
    const int wave = threadIdx.x >> 5;
    const int lane = threadIdx.x & 31;
    const int row  = blockIdx.x * 8 + wave;
    const float4* p = (const float4*)(X + (size_t)row * DIM);
    float4 v = p[lane];
    float  s = v.x * v.x + v.y * v.y + v.z * v.z + v.w * v.w;
#pragma unroll
    for (int m = 16; m >= 1; m >>= 1)
        s += __shfl_xor(s, m, 32);
    if (lane == 0) sq[row] = s;
}

// ---------------------------------------------------------------------------
// Kernel B: fused Gram (f32 WMMA) + hardest-pos / hardest-neg partials.
// Block (mb, jg): 8 waves; wave w owns M-tile rows mb*128 + w*16 .. +15.
// All waves share a double-buffered LDS panel of 2 j-tiles (32 rows x 128).
// ---------------------------------------------------------------------------
__global__ __launch_bounds__(THREADS) void triplet_partial_kernel(
    const float* __restrict__ X, const int* __restrict__ tgt,
    const float* __restrict__ sq,
    float* __restrict__ pmax, float* __restrict__ pmin) {
    __shared__ float Bs[2][BUF_FLOATS];

    const int tid  = threadIdx.x;
    const int wave = tid >> 5;
    const int lane = tid & 31;
    const int half = lane >> 4;     // K-pair selector within a 4-wide K step
    const int l16  = lane & 15;     // M index (A) / N index (B,C)
    const int jg   = blockIdx.y;
    const int i0   = blockIdx.x * (WAVES * TILE) + wave * TILE;
    const int jt0  = jg * TILES_PER_GROUP;     // first j-tile of this block

    // ---- A fragments for K = 128 (32 steps of 4), resident in VGPRs -------
    v2f afrag[32];
    const float* arow = X + (size_t)(i0 + l16) * DIM + half * 2;
#pragma unroll
    for (int s = 0; s < 32; ++s)
        afrag[s] = *(const v2f*)(arow + 4 * s);

    float sqi_r[8];
    int   tgt_r[8];
#pragma unroll
    for (int v = 0; v < 8; ++v) {
        const int m = v + 8 * half;       // C row held in VGPR v for this lane
        sqi_r[v] = sq[i0 + m];
        tgt_r[v] = tgt[i0 + m];
    }

    float maxp[8], minn[8];
#pragma unroll
    for (int v = 0; v < 8; ++v) { maxp[v] = -BIG; minn[v] = BIG; }

    // ---- cooperative staging of a 32-row B panel --------------------------
    // panel g covers rows (jt0 + 2g)*16 .. +31 ; thread copies 4 float4s.
#if USE_ASYNC_COPY
    auto stage_async = [&](int buf, int g) {
        const int j0 = (jt0 + 2 * g) * TILE;
#pragma unroll
        for (int k = 0; k < 4; ++k) {
            const int idx = tid + k * THREADS;      // float4 index 0..1023
            const int row = idx >> 5;               // 32 float4 per row
            const int c4  = idx & 31;
            const float* gsrc = X + (size_t)(j0 + row) * DIM + c4 * 4;
            float* ldst = &Bs[buf][row * BROW_STRIDE + c4 * 4];
            // AS1 pointer shares the generic bit pattern; AS3 pointer is the
            // low 32 bits of the generic LDS address (ISA 10.2 aperture map).
            __builtin_amdgcn_global_load_async_to_lds_b128(
                (as1_v4i*)(uintptr_t)gsrc,
                (as3_v4i*)(uint32_t)(uintptr_t)ldst, 0, 0);
        }
    };
#else
    auto stage_sync = [&](int buf, int g) {
        const int j0 = (jt0 + 2 * g) * TILE;
        float4 tmp[4];
#pragma unroll
        for (int k = 0; k < 4; ++k) {
            const int idx = tid + k * THREADS;
            const int row = idx >> 5;
            const int c4  = idx & 31;
            tmp[k] = *(const float4*)(X + (size_t)(j0 + row) * DIM + c4 * 4);
        }
#pragma unroll
        for (int k = 0; k < 4; ++k) {
            const int idx = tid + k * THREADS;
            const int row = idx >> 5;
            const int c4  = idx & 31;
            *(float4*)&Bs[buf][row * BROW_STRIDE + c4 * 4] = tmp[k];
        }
    };
#endif

    // ---- compute one staged panel (two j-tiles) ---------------------------
    auto compute = [&](int buf, int g) {
        const int j0 = (jt0 + 2 * g) * TILE;
        const float sqj0  = sq[j0 + l16];
        const float sqj1  = sq[j0 + 16 + l16];
        const int   tgtj0 = tgt[j0 + l16];
        const int   tgtj1 = tgt[j0 + 16 + l16];

        const float* b0base = &Bs[buf][l16 * BROW_STRIDE + half * 2];
        const float* b1base = b0base + 16 * BROW_STRIDE;

        v8f c0 = {}, c1 = {};
#pragma unroll
        for (int s = 0; s < 32; ++s) {
            v2f b0 = *(const v2f*)(b0base + 4 * s);
            v2f b1 = *(const v2f*)(b1base + 4 * s);
            c0 = __builtin_amdgcn_wmma_f32_16x16x4_f32(
                     false, afrag[s], false, b0, (short)0, c0, false, false);
            c1 = __builtin_amdgcn_wmma_f32_16x16x4_f32(
                     false, afrag[s], false, b1, (short)0, c1, false, false);
        }
#pragma unroll
        for (int v = 0; v < 8; ++v) {
            float d0 = fmaxf(sqi_r[v] + sqj0 - 2.0f * c0[v], 0.0f);
            float d1 = fmaxf(sqi_r[v] + sqj1 - 2.0f * c1[v], 0.0f);
            const bool s0 = (tgt_r[v] == tgtj0);
            const bool s1 = (tgt_r[v] == tgtj1);
            maxp[v] = s0 ? fmaxf(maxp[v], d0) : maxp[v];
            minn[v] = s0 ? minn[v] : fminf(minn[v], d0);
            maxp[v] = s1 ? fmaxf(maxp[v], d1) : maxp[v];
            minn[v] = s1 ? minn[v] : fminf(minn[v], d1);
        }
    };

    // ---- main pipeline ----------------------------------------------------
#if USE_ASYNC_COPY
    stage_async(0, 0);
    for (int it = 0; it < ITER; ++it) {
        const int cur = it & 1;
        if (it + 1 < ITER) {
            stage_async((it + 1) & 1, it + 1);
            __builtin_amdgcn_s_wait_asynccnt(4);   // group `it` complete
        } else {
            __builtin_amdgcn_s_wait_asynccnt(0);
        }
        __syncthreads();          // all waves' copies for buf `cur` visible
        compute(cur, it);
        __syncthreads();          // done reading before buf is re-staged
    }
#else
    for (int it = 0; it < ITER; ++it) {
        stage_sync(0, it);
        __syncthreads();
        compute(0, it);
        __syncthreads();
    }
#endif

    // ---- reduce over the 16 N-lanes of each half --------------------------
#pragma unroll
    for (int v = 0; v < 8; ++v) {
        float mp = maxp[v], mn = minn[v];
#pragma unroll
        for (int m = 8; m >= 1; m >>= 1) {
            mp = fmaxf(mp, __shfl_xor(mp, m, 32));
            mn = fminf(mn, __shfl_xor(mn, m, 32));
        }
        maxp[v] = mp; minn[v] = mn;
    }
    // Waves own disjoint rows -> write partials directly (unique addresses).
    if (l16 == 0) {
#pragma unroll
        for (int v = 0; v < 8; ++v) {
            const int row = i0 + v + 8 * half;
            pmax[jg * N_ROWS + row] = maxp[v];
            pmin[jg * N_ROWS + row] = minn[v];
        }
    }
}

// ---------------------------------------------------------------------------
// Kernel C: combine j-group partials per row + deterministic fixed-order sum.
// ---------------------------------------------------------------------------
__global__ __launch_bounds__(256) void combine_sum_kernel(
    const float* __restrict__ pmax, const float* __restrict__ pmin,
    float* __restrict__ out) {
    __shared__ float buf[256];
    float s = 0.0f;
    for (int row = threadIdx.x; row < N_ROWS; row += 256) {
        float dp = -BIG, dn = BIG;
#pragma unroll
        for (int g = 0; g < JGROUPS; ++g) {
            dp = fmaxf(dp, pmax[g * N_ROWS + row]);
            dn = fminf(dn, pmin[g * N_ROWS + row]);
        }
        const float l = dp - dn + MARGIN;
        s += (l > 0.0f) ? l : 0.0f;
    }
    buf[threadIdx.x] = s;
    __syncthreads();
#pragma unroll
    for (int off = 128; off >= 1; off >>= 1) {
        if (threadIdx.x < off) buf[threadIdx.x] += buf[threadIdx.x + off];
        __syncthreads();
    }
    if (threadIdx.x == 0) out[0] = buf[0];
}

// ---------------------------------------------------------------------------
extern "C" void kernel_launch(void* const* d_in, const int* in_sizes, int n_in,
                              void* d_out, int out_size, void* d_ws, size_t ws_size,
                              hipStream_t stream) {
    const float* X   = (const float*)d_in[0];   // [8192, 128] f32
    const int*   tgt = (const int*)d_in[1];     // [8192] labels

    float* sq   = (float*)d_ws;                 // 8192 f32
    float* pmax = sq + N_ROWS;                  // 8 * 8192 f32
    float* pmin = pmax + JGROUPS * N_ROWS;      // 8 * 8192 f32

    sqnorm_kernel<<<N_ROWS / 8, 256, 0, stream>>>(X, sq);
    dim3 grid(N_ROWS / (WAVES * TILE), JGROUPS);    // (64, 8)
    triplet_partial_kernel<<<grid, THREADS, 0, stream>>>(X, tgt, sq, pmax, pmin);
    combine_sum_kernel<<<1, 256, 0, stream>>>(pmax, pmin, (float*)d_out);
}